// ASTPruner_86079734546533
// MI455X (gfx1250) — compile-verified
//
#include <hip/hip_runtime.h>
#include <hip/hip_bf16.h>
#include <cstddef>

#define B_    8
#define T_    32
#define N_    196
#define C_    768
#define TN_   (T_ * N_)        // 6272
#define K_SEL 3136             // int(0.5 * T * N)
#define EPSF  1e-6f

typedef float v2f __attribute__((ext_vector_type(2)));
typedef float v8f __attribute__((ext_vector_type(8)));

// ---------- small device helpers ----------
__device__ __forceinline__ float wred_max(float v) {
#pragma unroll
  for (int d = 16; d > 0; d >>= 1) v = fmaxf(v, __shfl_xor(v, d, 32));
  return v;
}
__device__ __forceinline__ float wred_min(float v) {
#pragma unroll
  for (int d = 16; d > 0; d >>= 1) v = fminf(v, __shfl_xor(v, d, 32));
  return v;
}
__device__ __forceinline__ float wred_sum(float v) {
#pragma unroll
  for (int d = 16; d > 0; d >>= 1) v += __shfl_xor(v, d, 32);
  return v;
}
__device__ __forceinline__ float sigmoidf(float x) {
  return 1.0f / (1.0f + __expf(-x));
}
// linear interp of H (length T_in) back to T_=32 at position t (reference _interp_linear_T)
__device__ __forceinline__ float interp_T(const float* H, int T_in, int t) {
  float scale = (float)T_in / (float)T_;
  float src = fmaxf(((float)t + 0.5f) * scale - 0.5f, 0.0f);
  int i0 = (int)floorf(src);
  int i1 = min(i0 + 1, T_in - 1);
  float w = src - (float)i0;
  return H[i0] * (1.0f - w) + H[i1] * w;
}

// ---------- kernel 1: region assignment (argmin over centers) + counts ----------
__global__ void setup_regions_kernel(const float* __restrict__ cc,   // (4,2)
                                     const float* __restrict__ cf,   // (8,2)
                                     const float* __restrict__ coords,// (196,2)
                                     int* __restrict__ rid_c, int* __restrict__ rid_f,
                                     float* __restrict__ cnt) {      // (16,)
  __shared__ int cntI[16];
  int tid = threadIdx.x;
  if (tid < 16) cntI[tid] = 0;
  __syncthreads();
  if (tid < N_) {
    float u = coords[2 * tid], v = coords[2 * tid + 1];
    int bc = 0; float bd = 3.4e38f;
#pragma unroll
    for (int r = 0; r < 4; r++) {
      float du = u - cc[2 * r], dv = v - cc[2 * r + 1];
      float d = du * du + dv * dv;
      if (d < bd) { bd = d; bc = r; }
    }
    rid_c[tid] = bc;
    atomicAdd(&cntI[bc], 1);
    int bf = 0; bd = 3.4e38f;
#pragma unroll
    for (int r = 0; r < 8; r++) {
      float du = u - cf[2 * r], dv = v - cf[2 * r + 1];
      float d = du * du + dv * dv;
      if (d < bd) { bd = d; bf = r; }
    }
    rid_f[tid] = bf;
    atomicAdd(&cntI[4 + bf], 1);
  }
  __syncthreads();
  if (tid < 16) cnt[tid] = (float)cntI[tid];
}

// ---------- kernel 2: temporal entropy, one workgroup per (b,n) ----------
// Full T x C softmax tile lives in LDS (96 KB -- CDNA5 WGP LDS).
__global__ __launch_bounds__(256) void temporal_kernel(const float* __restrict__ x,
                                                       float* __restrict__ Htime) {
  __shared__ float p[T_][C_];              // 96 KB
  __shared__ float H1[T_], H2[T_], H4[T_];
  int bn = blockIdx.x;
  int b = bn / N_, n = bn - b * N_;
  int tid = threadIdx.x, lane = tid & 31, w = tid >> 5;

  // phase 1: softmax + entropy(p) per row (window L=1)
  for (int t = w; t < T_; t += 8) {
    const float* row = x + (((size_t)(b * T_ + t)) * N_ + n) * C_;
    float loc[24];
    float m = -3.4e38f;
#pragma unroll
    for (int i = 0; i < 24; i++) { loc[i] = row[lane + 32 * i]; m = fmaxf(m, loc[i]); }
    m = wred_max(m);
    float s = 0.0f;
#pragma unroll
    for (int i = 0; i < 24; i++) { loc[i] = __expf(loc[i] - m); s += loc[i]; }
    s = wred_sum(s);
    float inv = 1.0f / s;
    float ent = 0.0f;
#pragma unroll
    for (int i = 0; i < 24; i++) {
      float pv = loc[i] * inv;
      p[t][lane + 32 * i] = pv;
      ent -= pv * __logf(pv + EPSF);
    }
    ent = wred_sum(ent);
    if (lane == 0) H1[t] = ent;
  }
  __syncthreads();

  // phase 2: window L=2 (T_in=31)
  for (int t = w; t < T_ - 1; t += 8) {
    float ent = 0.0f;
#pragma unroll
    for (int i = 0; i < 24; i++) {
      int c = lane + 32 * i;
      float q = 0.5f * (p[t][c] + p[t + 1][c]);
      ent -= q * __logf(q + EPSF);
    }
    ent = wred_sum(ent);
    if (lane == 0) H2[t] = ent;
  }
  // window L=4 (T_in=29)
  for (int t = w; t < T_ - 3; t += 8) {
    float ent = 0.0f;
#pragma unroll
    for (int i = 0; i < 24; i++) {
      int c = lane + 32 * i;
      float q = 0.25f * (p[t][c] + p[t + 1][c] + p[t + 2][c] + p[t + 3][c]);
      ent -= q * __logf(q + EPSF);
    }
    ent = wred_sum(ent);
    if (lane == 0) H4[t] = ent;
  }
  __syncthreads();

  if (tid < T_) {
    int t = tid;
    float h = H1[t] + interp_T(H2, T_ - 1, t) + interp_T(H4, T_ - 3, t);
    Htime[((size_t)(b * T_ + t)) * N_ + n] = h * (1.0f / 3.0f);
  }
}

// ---------- kernel 3: region entropies via V_WMMA_F32_16X16X4_F32 ----------
// One workgroup per (b,t). D(16x16) = A(16x4 one-hot regions) x B(4 tokens x 16 channels),
// accumulated over 49 chunks of 4 tokens. Rows 0-3 = coarse regions, 4-11 = fine, 12-15 = 0.
__global__ __launch_bounds__(256) void region_wmma_kernel(const float* __restrict__ x,
                                                          const int* __restrict__ rid_cg,
                                                          const int* __restrict__ rid_fg,
                                                          const float* __restrict__ cntg,
                                                          float* __restrict__ Hc,
                                                          float* __restrict__ Hf) {
  __shared__ float p4[4][C_];   // 12 KB: softmax of 4 token rows
  __shared__ int   rc[N_], rf[N_];
  __shared__ float Hreg[16];
  __shared__ float cnt[16];

  int bt = blockIdx.x;
  int b = bt / T_, t = bt - b * T_;
  int tid = threadIdx.x, lane = tid & 31, w = tid >> 5;
  int m = lane & 15;

  for (int i = tid; i < N_; i += 256) { rc[i] = rid_cg[i]; rf[i] = rid_fg[i]; }
  if (tid < 16) { cnt[tid] = cntg[tid]; Hreg[tid] = 0.0f; }

  v8f acc[6];
#pragma unroll
  for (int jj = 0; jj < 6; jj++) acc[jj] = (v8f){0, 0, 0, 0, 0, 0, 0, 0};

  for (int kb = 0; kb < 49; kb++) {
    __syncthreads();   // p4 free for rewrite; also makes rc/rf/cnt visible on kb==0
    if (w < 4) {       // waves 0-3: softmax of one token row each
      int tok = kb * 4 + w;
      const float* row = x + (((size_t)(b * T_ + t)) * N_ + tok) * C_;
      float loc[24];
      float mx = -3.4e38f;
#pragma unroll
      for (int i = 0; i < 24; i++) { loc[i] = row[lane + 32 * i]; mx = fmaxf(mx, loc[i]); }
      mx = wred_max(mx);
      float s = 0.0f;
#pragma unroll
      for (int i = 0; i < 24; i++) { loc[i] = __expf(loc[i] - mx); s += loc[i]; }
      s = wred_sum(s);
      float inv = 1.0f / s;
#pragma unroll
      for (int i = 0; i < 24; i++) p4[w][lane + 32 * i] = loc[i] * inv;
    }
    __syncthreads();

    // A: 16x4 one-hot. ISA layout: lane m holds row m; VGPR0 = {K0 | K2}, VGPR1 = {K1 | K3}.
    int kk = (lane >> 4) << 1;           // 0 for lanes 0-15, 2 for lanes 16-31
    int tok0 = kb * 4 + kk;
    int r0c = rc[tok0], r0f = rf[tok0];
    int r1c = rc[tok0 + 1], r1f = rf[tok0 + 1];
    float ax = (m < 4) ? (r0c == m ? 1.0f : 0.0f)
                       : ((m < 12) ? (r0f == m - 4 ? 1.0f : 0.0f) : 0.0f);
    float ay = (m < 4) ? (r1c == m ? 1.0f : 0.0f)
                       : ((m < 12) ? (r1f == m - 4 ? 1.0f : 0.0f) : 0.0f);
    v2f A = {ax, ay};

    int prow = kk;                       // B mirrors A: VGPR0 = rows {0|2}, VGPR1 = rows {1|3}
#pragma unroll
    for (int jj = 0; jj < 6; jj++) {
      int col = (w * 6 + jj) * 16 + m;
      v2f Bv = {p4[prow][col], p4[prow + 1][col]};
      acc[jj] = __builtin_amdgcn_wmma_f32_16x16x4_f32(
          false, A, false, Bv, (short)0, acc[jj], false, false);
    }
  }
  __syncthreads();

  // D layout: VGPR i -> (M=i, N=lane) for lanes 0-15, (M=i+8, N=lane-16) for lanes 16-31.
#pragma unroll
  for (int jj = 0; jj < 6; jj++) {
#pragma unroll
    for (int i = 0; i < 8; i++) {
      int r = (lane < 16) ? i : (i + 8);
      if (r < 12) {
        float v = acc[jj][i] / (cnt[r] + EPSF);
        atomicAdd(&Hreg[r], -v * __logf(v + EPSF));
      }
    }
  }
  __syncthreads();
  if (tid < 4)       Hc[bt * 4 + tid] = Hreg[tid];
  else if (tid < 12) Hf[bt * 8 + (tid - 4)] = Hreg[tid];
}

// ---------- kernel 4: normalize, score, exact k-th largest, mask ----------
__device__ void block_minmax(const float* __restrict__ src, int count,
                             float* redA, float* redB, float* out2, int tid, int nthreads) {
  float mn = 3.4e38f, mx = -3.4e38f;
  for (int i = tid; i < count; i += nthreads) {
    float v = src[i];
    mn = fminf(mn, v);
    mx = fmaxf(mx, v);
  }
  mn = wred_min(mn);
  mx = wred_max(mx);
  int lane = tid & 31, w = tid >> 5;
  if (lane == 0) { redA[w] = mn; redB[w] = mx; }
  __syncthreads();
  if (w == 0) {
    int nw = (nthreads + 31) >> 5;
    float a = (lane < nw) ? redA[lane] : 3.4e38f;
    float c = (lane < nw) ? redB[lane] : -3.4e38f;
    a = wred_min(a);
    c = wred_max(c);
    if (lane == 0) { out2[0] = a; out2[1] = c; }
  }
  __syncthreads();
}

__global__ __launch_bounds__(1024) void score_kernel(const float* __restrict__ Htime,
                                                     const float* __restrict__ Hc,
                                                     const float* __restrict__ Hf,
                                                     const int* __restrict__ rid_cg,
                                                     const int* __restrict__ rid_fg,
                                                     float* __restrict__ out) {
  __shared__ float sc[TN_];          // 25 KB
  __shared__ float redA[32], redB[32];
  __shared__ float st[6];            // mnT mxT mnC mxC mnF mxF
  __shared__ int   rc[N_], rf[N_];
  __shared__ float kval;
  int b = blockIdx.x, tid = threadIdx.x;

  for (int i = tid; i < N_; i += 1024) { rc[i] = rid_cg[i]; rf[i] = rid_fg[i]; }

  block_minmax(Htime + (size_t)b * TN_, TN_, redA, redB, &st[0], tid, 1024);
  block_minmax(Hc + b * (T_ * 4), T_ * 4, redA, redB, &st[2], tid, 1024);
  block_minmax(Hf + b * (T_ * 8), T_ * 8, redA, redB, &st[4], tid, 1024);

  float mnT = st[0], iT = 1.0f / (st[1] - st[0] + EPSF);
  float mnC = st[2], iC = 1.0f / (st[3] - st[2] + EPSF);
  float mnF = st[4], iF = 1.0f / (st[5] - st[4] + EPSF);

  for (int idx = tid; idx < TN_; idx += 1024) {
    int t = idx / N_, n = idx - t * N_;
    float sT = (Htime[(size_t)b * TN_ + idx] - mnT) * iT;
    float sC = (Hc[b * (T_ * 4) + t * 4 + rc[n]] - mnC) * iC;
    float sF = (Hf[b * (T_ * 8) + t * 8 + rf[n]] - mnF) * iF;
    sc[idx] = sT + 0.5f * sC + 0.5f * sF;   // ALPHA=1, BETA=GAMMA=0.5
  }
  __syncthreads();

  // exact k-th largest (desc index K_SEL-1), tie-correct rank count
  for (int idx = tid; idx < TN_; idx += 1024) {
    float v = sc[idx];
    int gt = 0, eq = 0;
    for (int j = 0; j < TN_; j++) {
      float u = sc[j];
      gt += (u > v);
      eq += (u == v);
    }
    if (gt <= K_SEL - 1 && gt + eq >= K_SEL) kval = v;  // all qualifiers share one value
  }
  __syncthreads();
  float kv = kval;
  for (int idx = tid; idx < TN_; idx += 1024) {
    out[(size_t)b * TN_ + idx] = sigmoidf((sc[idx] - kv) * 10.0f);  // /TEMP, TEMP=0.1
  }
}

// ---------- kernel 5: tail outputs (head_w, ch_w, block_w, L_AST) ----------
__global__ void tail_kernel(const float* __restrict__ gh, const float* __restrict__ gc,
                            const float* __restrict__ gb, float* __restrict__ out) {
  const int NH = 144, NC = 36864, NB = 12;
  const int OFF_H = B_ * TN_;               // 50176
  int i = blockIdx.x * blockDim.x + threadIdx.x;
  if (i < NH)                     out[OFF_H + i] = sigmoidf(gh[i]);
  else if (i < NH + NC)           out[OFF_H + NH + (i - NH)] = sigmoidf(gc[i - NH]);
  else if (i < NH + NC + NB)      out[OFF_H + NH + NC + (i - NH - NC)] = sigmoidf(gb[i - NH - NC]);
  else if (i == NH + NC + NB)     out[OFF_H + NH + NC + NB] = 0.0f;  // L_AST: all LAM_* == 0
}

// ---------- launch ----------
extern "C" void kernel_launch(void* const* d_in, const int* in_sizes, int n_in,
                              void* d_out, int out_size, void* d_ws, size_t ws_size,
                              hipStream_t stream) {
  const float* x       = (const float*)d_in[0];  // (8,32,196,768)
  const float* cc      = (const float*)d_in[1];  // (4,2)
  const float* cf      = (const float*)d_in[2];  // (8,2)
  const float* g_head  = (const float*)d_in[3];  // (12,12)
  const float* g_ch    = (const float*)d_in[4];  // (12,3072)
  const float* g_block = (const float*)d_in[5];  // (12,)
  const float* coords  = (const float*)d_in[6];  // (196,2)
  float* out = (float*)d_out;

  char* wsb = (char*)d_ws;
  int*   rid_c = (int*)(wsb + 0);                          // 196 ints (1 KB slot)
  int*   rid_f = (int*)(wsb + 1024);                       // 196 ints (1 KB slot)
  float* cnt   = (float*)(wsb + 2048);                     // 16 floats
  float* Htime = (float*)(wsb + 4096);                     // 50176 floats
  float* Hc    = (float*)(wsb + 4096 + 200704);            // 1024 floats
  float* Hf    = (float*)(wsb + 4096 + 200704 + 4096);     // 2048 floats

  setup_regions_kernel<<<1, 256, 0, stream>>>(cc, cf, coords, rid_c, rid_f, cnt);
  temporal_kernel<<<B_ * N_, 256, 0, stream>>>(x, Htime);
  region_wmma_kernel<<<B_ * T_, 256, 0, stream>>>(x, rid_c, rid_f, cnt, Hc, Hf);
  score_kernel<<<B_, 1024, 0, stream>>>(Htime, Hc, Hf, rid_c, rid_f, out);
  tail_kernel<<<(144 + 36864 + 12 + 1 + 255) / 256, 256, 0, stream>>>(g_head, g_ch, g_block, out);
}